// T2T_module_90452011254286
// MI455X (gfx1250) — compile-verified
//
#include <hip/hip_runtime.h>
#include <hip/hip_bf16.h>
#include <math.h>

// ---------------------------------------------------------------------------
// T2T-ViT stem (two Performer blocks) for MI455X / gfx1250, wave32.
// All GEMMs on V_WMMA_F32_16X16X4_F32. Hot loops are guard-free:
//   M % 16 == 0, N % 32 == 0, K % 8 == 0 for every call (K padded with zeros).
// ---------------------------------------------------------------------------

typedef float v2f __attribute__((ext_vector_type(2)));
typedef float v8f __attribute__((ext_vector_type(8)));

enum { EPI_NONE = 0, EPI_PRMEXP = 1, EPI_DIVD = 2, EPI_GELU = 3, EPI_ADD = 4 };

__device__ __forceinline__ float gelu_exact(float v) {
    return 0.5f * v * (1.0f + erff(v * 0.70710678118654752f));
}

__device__ __forceinline__ v8f wmma4(v2f a, v2f b, v8f c) {
    return __builtin_amdgcn_wmma_f32_16x16x4_f32(false, a, false, b, (short)0, c,
                                                 false, false);
}

// ---------------------------------------------------------------------------
// Strided batched GEMM: C[b] = epi(A[b](MxK) * B[b](KxN) + bias)
// One wave -> 16(M) x 32(N) strip = 2 WMMA accumulators; K-loop step 8 (x2 unroll).
//
// A fragment (16x4 f32): lane holds A[m0+(lane&15)][k0+2*(lane>>4)+{0,1}]
// B fragment (4x16 f32): lane holds B[k0+2*(lane>>4)+{0,1}][n0+(lane&15)]
// C tile (16x16 f32):   VGPR r of lane L = C[m0+r+8*(L>>4)][n0+(L&15)]
//
// AM: 1 = A pair contiguous (csA==1), 0 = generic strided
// BM: 2 = packed pair layout Bp[((k/4)*2+h)*N + n] = {B[k..],B[k+1..]},
//     1 = pair contiguous (rsB==1), 0 = generic strided
// ---------------------------------------------------------------------------
template <int AM, int BM>
__global__ __launch_bounds__(256) void gemm_wmma_kernel(
    const float* __restrict__ A, long rsA, long csA, long bsA,
    const float* __restrict__ Bm, long rsB, long csB, long bsB,
    const float* __restrict__ bias,
    const float* __restrict__ rv, long rvStride,
    const float* __restrict__ Add, long rsAdd, long bsAdd,
    float* __restrict__ C, long rsC, long bsC,
    int M, int N, int K, int epi, float scale)
{
    const int lane = threadIdx.x & 31;
    const int wave = threadIdx.x >> 5;
    const long gw = (long)blockIdx.x * 8 + wave;
    const long tilesN = (long)N >> 5;             // 32-column strips
    const long tileM = gw / tilesN;
    const long tileN = gw - tileM * tilesN;
    if (tileM * 16 >= (long)M) return;            // wave-uniform exit

    const int b  = blockIdx.z;
    const int m0 = (int)tileM * 16;
    const int n0 = (int)tileN * 32;
    const int h  = lane >> 4;                     // 0/1: K-half select
    const int kH = h << 1;
    const int nl = lane & 15;
    const int n1 = n0 + nl;
    const int n2 = n1 + 16;
    const int mA = m0 + nl;

    // ---- per-lane base pointers (no address math in the loop) ----
    const float* pa = A + (long)b * bsA + (long)mA * rsA + (long)kH * csA;
    const long aStep = 4 * csA;

    const float *pb1, *pb2;
    long bStep;
    if (BM == 2) {          // packed: idx = ((k/4)*2 + h)*N + n, float2 elements
        pb1 = Bm + (((long)h * N) + n1) * 2;
        pb2 = pb1 + 32;     // +16 columns * 2 floats
        bStep = (long)4 * N;
    } else {
        pb1 = Bm + (long)b * bsB + (long)kH * rsB + (long)n1 * csB;
        pb2 = Bm + (long)b * bsB + (long)kH * rsB + (long)n2 * csB;
        bStep = 4 * rsB;
    }

    v8f z = {0.f, 0.f, 0.f, 0.f, 0.f, 0.f, 0.f, 0.f};
    v8f acc0 = z, acc1 = z;

    for (int k0 = 0; k0 < K; k0 += 8) {
        v2f av0, av1, b10, b20, b11, b21;
        if (AM == 1) {
            av0 = *(const v2f*)pa;
            av1 = *(const v2f*)(pa + aStep);
        } else {
            av0[0] = pa[0];         av0[1] = pa[csA];
            av1[0] = pa[aStep];     av1[1] = pa[aStep + csA];
        }
        if (BM >= 1) {              // packed or rsB==1: pair is contiguous
            b10 = *(const v2f*)pb1;           b20 = *(const v2f*)pb2;
            b11 = *(const v2f*)(pb1 + bStep); b21 = *(const v2f*)(pb2 + bStep);
        } else {
            b10[0] = pb1[0];         b10[1] = pb1[rsB];
            b20[0] = pb2[0];         b20[1] = pb2[rsB];
            b11[0] = pb1[bStep];     b11[1] = pb1[bStep + rsB];
            b21[0] = pb2[bStep];     b21[1] = pb2[bStep + rsB];
        }
        acc0 = wmma4(av0, b10, acc0);
        acc1 = wmma4(av0, b20, acc1);
        acc0 = wmma4(av1, b11, acc0);
        acc1 = wmma4(av1, b21, acc1);
        pa  += 2 * aStep;
        pb1 += 2 * bStep;
        pb2 += 2 * bStep;
    }

    // ---- epilogue ----
    const float* rvp = rv ? rv + (long)b * rvStride : nullptr;
    float* Cb = C + (long)b * bsC;
    float bias1 = 0.f, bias2 = 0.f;
    if (bias) { bias1 = bias[n1]; bias2 = bias[n2]; }
#pragma unroll
    for (int r = 0; r < 8; ++r) {
        const int row = m0 + r + 8 * h;
        float v1 = acc0[r] + bias1;
        float v2 = acc1[r] + bias2;
        if (epi == EPI_PRMEXP) {
            const float xd = rvp[row];
            v1 = expf(v1 - xd) * scale;
            v2 = expf(v2 - xd) * scale;
        } else if (epi == EPI_DIVD) {
            const float dnm = 1.0f / (rvp[row] + 1e-8f);
            v1 *= dnm; v2 *= dnm;
        } else if (epi == EPI_GELU) {
            v1 = gelu_exact(v1); v2 = gelu_exact(v2);
        } else if (epi == EPI_ADD) {
            const float* ap = Add + (long)b * bsAdd + (long)row * rsAdd;
            v1 += ap[n1]; v2 += ap[n2];
        }
        const long ro = (long)row * rsC;
        Cb[ro + n1] = v1;
        Cb[ro + n2] = v2;
    }
}

// ---------------------------------------------------------------------------
// Pack weight B (K x N row-major) into WMMA pair layout with zero K-padding:
//   Bp[ ((g*2+h)*N + n)*2 + e ] = (4g+2h+e < K) ? W[(4g+2h+e)*N + n] : 0
// ---------------------------------------------------------------------------
__global__ void pack_b_kernel(const float* __restrict__ W, float* __restrict__ Bp,
                              int K, int N, int Kpad)
{
    const long total = (long)(Kpad >> 2) * 2 * N;
    const long idx = (long)blockIdx.x * blockDim.x + threadIdx.x;
    if (idx >= total) return;
    const int n = (int)(idx % N);
    const long gh = idx / N;
    const int hh = (int)(gh & 1);
    const long g = gh >> 1;
    const int k = (int)(g * 4 + hh * 2);
    const float e0 = (k < K)     ? W[(long)k * N + n]       : 0.f;
    const float e1 = (k + 1 < K) ? W[(long)(k + 1) * N + n] : 0.f;
    Bp[idx * 2]     = e0;
    Bp[idx * 2 + 1] = e1;
}

// ---------------------------------------------------------------------------
// im2col / unfold, generalized input strides + output row stride.
// Feature order f = (c*k + kh)*k + kw.
// ---------------------------------------------------------------------------
__global__ void unfold_kernel(const float* __restrict__ in, float* __restrict__ out,
                              int Bn, int Cin, int H, int W, int k, int s, int p,
                              int Hout, int Wout, long strB, long strC, long strHW,
                              long ldOut)
{
    const int Cf = Cin * k * k;
    const long T = (long)Hout * Wout;
    const long total = (long)Bn * T * Cf;
    const long idx = (long)blockIdx.x * blockDim.x + threadIdx.x;
    if (idx >= total) return;
    const int f = (int)(idx % Cf);
    const long bt = idx / Cf;
    const int t = (int)(bt % T);
    const int b = (int)(bt / T);
    const int kw = f % k;
    const int kh = (f / k) % k;
    const int c = f / (k * k);
    const int ow = t % Wout;
    const int oh = t / Wout;
    const int y = oh * s - p + kh;
    const int x = ow * s - p + kw;
    float v = 0.f;
    if (y >= 0 && y < H && x >= 0 && x < W)
        v = in[(long)b * strB + (long)c * strC + ((long)y * W + x) * strHW];
    out[bt * ldOut + f] = v;
}

// ---------------------------------------------------------------------------
// LayerNorm, one wave32 per row; zero-fills pad columns [d, dPad).
// In-place safe when out==in.
// ---------------------------------------------------------------------------
__global__ void ln_kernel(const float* __restrict__ in, long inStride,
                          float* __restrict__ out, long outStride,
                          const float* __restrict__ g, const float* __restrict__ b,
                          long rows, int d, int dPad)
{
    const int lane = threadIdx.x & 31;
    const long row = ((long)blockIdx.x * blockDim.x + threadIdx.x) >> 5;
    if (row >= rows) return;
    const float* x = in + row * inStride;
    float s = 0.f, ss = 0.f;
    for (int i = lane; i < d; i += 32) { float v = x[i]; s += v; ss += v * v; }
#pragma unroll
    for (int o = 16; o > 0; o >>= 1) {
        s  += __shfl_xor(s,  o, 32);
        ss += __shfl_xor(ss, o, 32);
    }
    const float mu = s / (float)d;
    const float var = ss / (float)d - mu * mu;
    const float rstd = rsqrtf(var + 1e-5f);
    float* y = out + row * outStride;
    for (int i = lane; i < d; i += 32) y[i] = (x[i] - mu) * rstd * g[i] + b[i];
    for (int i = d + lane; i < dPad; i += 32) y[i] = 0.f;
}

// xd_k[r] = 0.5*||K[r]||^2 , xd_q[r] = 0.5*||Q[r]||^2 from KQV rows (stride 192)
__global__ void rowhalfnorm_kernel(const float* __restrict__ KQV,
                                   float* __restrict__ xdk, float* __restrict__ xdq,
                                   long rows)
{
    const long r = (long)blockIdx.x * blockDim.x + threadIdx.x;
    if (r >= rows) return;
    const float* p = KQV + r * 192;
    float sk = 0.f, sq = 0.f;
#pragma unroll 8
    for (int i = 0; i < 64; ++i) {
        float a = p[i];      sk += a * a;
        float q = p[64 + i]; sq += q * q;
    }
    xdk[r] = 0.5f * sk;
    xdq[r] = 0.5f * sq;
}

// kps[b,:] = sum_t KP[b,t,:]  (32 columns), one block per batch
__global__ void kpsum_kernel(const float* __restrict__ KP, float* __restrict__ kps, int T)
{
    __shared__ float sm[256];
    const int b = blockIdx.x;
    const int tid = threadIdx.x;
    const int col = tid & 31, rg = tid >> 5;
    const float* base = KP + (long)b * T * 32;
    float s = 0.f;
    for (int t = rg; t < T; t += 8) s += base[(long)t * 32 + col];
    sm[tid] = s;
    __syncthreads();
    if (tid < 32) {
        float tot = 0.f;
#pragma unroll
        for (int j = 0; j < 8; ++j) tot += sm[j * 32 + tid];
        kps[b * 32 + tid] = tot;
    }
}

// D[r] = dot(QP[r,:], kps[batch(r),:])
__global__ void dvec_kernel(const float* __restrict__ QP, const float* __restrict__ kps,
                            float* __restrict__ Dv, long rows, int T)
{
    const long r = (long)blockIdx.x * blockDim.x + threadIdx.x;
    if (r >= rows) return;
    const int b = (int)(r / T);
    const float* q = QP + r * 32;
    const float* s = kps + b * 32;
    float acc = 0.f;
#pragma unroll
    for (int m = 0; m < 32; ++m) acc += q[m] * s[m];
    Dv[r] = acc;
}

// ---------------------------------------------------------------------------
// Host-side helpers
// ---------------------------------------------------------------------------
static void launch_gemm(hipStream_t st, int AM, int BM,
    const float* A, long rsA, long csA, long bsA,
    const float* Bm, long rsB, long csB, long bsB,
    const float* bias, const float* rv, long rvStride,
    const float* Add, long rsAdd, long bsAdd,
    float* C, long rsC, long bsC,
    int M, int N, int K, int batch, int epi, float scale)
{
    const long waves = ((long)M / 16) * ((long)N / 32);
    const int blocks = (int)((waves + 7) / 8);
    dim3 grid(blocks, 1, batch);
#define GCALL(am, bm) gemm_wmma_kernel<am, bm><<<grid, 256, 0, st>>>( \
        A, rsA, csA, bsA, Bm, rsB, csB, bsB, bias, rv, rvStride,      \
        Add, rsAdd, bsAdd, C, rsC, bsC, M, N, K, epi, scale)
    if      (AM == 1 && BM == 2) GCALL(1, 2);
    else if (AM == 1 && BM == 1) GCALL(1, 1);
    else if (AM == 0 && BM == 0) GCALL(0, 0);
    else                         GCALL(1, 0);
#undef GCALL
}

static void launch_pack(hipStream_t st, const float* W, float* Bp, int K, int N, int Kpad)
{
    const long total = (long)(Kpad >> 2) * 2 * N;
    pack_b_kernel<<<(int)((total + 255) / 256), 256, 0, st>>>(W, Bp, K, N, Kpad);
}

static void launch_ln(hipStream_t st, const float* in, long inStride, float* out,
                      long outStride, const float* g, const float* b,
                      long rows, int d, int dPad)
{
    const long threads = rows * 32;
    ln_kernel<<<(int)((threads + 255) / 256), 256, 0, st>>>(
        in, inStride, out, outStride, g, b, rows, d, dPad);
}

struct PerfW {
    const float *kqv_w, *kqv_b, *proj_w, *proj_b;
    const float *ln1_g, *ln1_b, *ln2_g, *ln2_b;
    const float *m1_w, *m1_b, *m2_w, *m2_b, *rf;
};

struct Regions {
    float *X, *KQV, *KP, *QP, *xdk, *xdq, *Dv, *kps, *kptv;
    float *yatt, *y, *yn, *g, *h;
    float *pKQV, *pP, *pM1, *pM2;   // packed weights
};

static void run_performer(hipStream_t st, const Regions& R, const PerfW& P,
                          int Bn, int T, int d, int dPad)
{
    const long Mr = (long)Bn * T;
    const int M = (int)Mr;
    const float invSqrtM = 0.17677669529663687f;  // 1/sqrt(32)

    // pack this block's weights (zero-filled to dPad where needed)
    launch_pack(st, P.kqv_w, R.pKQV, d, 192, dPad);
    launch_pack(st, P.proj_w, R.pP, 64, 64, 64);
    launch_pack(st, P.m1_w, R.pM1, 64, 64, 64);
    launch_pack(st, P.m2_w, R.pM2, 64, 64, 64);

    // LN1 in place on padded patches (zero-fills pad cols)
    launch_ln(st, R.X, dPad, R.X, dPad, P.ln1_g, P.ln1_b, Mr, d, dPad);

    // KQV = XN @ kqv_w + b    (M x dPad) @ (dPad x 192)
    launch_gemm(st, 1, 2, R.X, dPad, 1, 0, R.pKQV, 0, 0, 0, P.kqv_b,
                nullptr, 0, nullptr, 0, 0, R.KQV, 192, 0,
                M, 192, dPad, 1, EPI_NONE, 0.f);

    rowhalfnorm_kernel<<<(int)((Mr + 255) / 256), 256, 0, st>>>(R.KQV, R.xdk, R.xdq, Mr);

    // KP = exp(K @ rf^T - xd_k)/sqrt(m):  B(k,n) = rf[n*64+k] -> rsB=1, csB=64
    launch_gemm(st, 1, 1, R.KQV + 0, 192, 1, 0, P.rf, 1, 64, 0, nullptr,
                R.xdk, 0, nullptr, 0, 0, R.KP, 32, 0,
                M, 32, 64, 1, EPI_PRMEXP, invSqrtM);
    launch_gemm(st, 1, 1, R.KQV + 64, 192, 1, 0, P.rf, 1, 64, 0, nullptr,
                R.xdq, 0, nullptr, 0, 0, R.QP, 32, 0,
                M, 32, 64, 1, EPI_PRMEXP, invSqrtM);

    kpsum_kernel<<<Bn, 256, 0, st>>>(R.KP, R.kps, T);
    dvec_kernel<<<(int)((Mr + 255) / 256), 256, 0, st>>>(R.QP, R.kps, R.Dv, Mr, T);

    // kptv[b] (64x32) = V[b]^T (64xT) @ KP[b] (Tx32); V = KQV[:,128:192]
    launch_gemm(st, 0, 0, R.KQV + 128, 1, 192, (long)T * 192,
                R.KP, 32, 1, (long)T * 32, nullptr,
                nullptr, 0, nullptr, 0, 0, R.kptv, 32, 64 * 32,
                64, 32, T, Bn, EPI_NONE, 0.f);

    // y_att[b] (Tx64) = QP[b] (Tx32) @ kptv[b]^T, / (D + 1e-8)
    launch_gemm(st, 1, 1, R.QP, 32, 1, (long)T * 32,
                R.kptv, 1, 32, 64 * 32, nullptr,
                R.Dv, T, nullptr, 0, 0, R.yatt, 64, (long)T * 64,
                T, 64, 32, Bn, EPI_DIVD, 0.f);

    // y = V + (y_att @ proj_w + proj_b)
    launch_gemm(st, 1, 2, R.yatt, 64, 1, 0, R.pP, 0, 0, 0, P.proj_b,
                nullptr, 0, R.KQV + 128, 192, 0, R.y, 64, 0,
                M, 64, 64, 1, EPI_ADD, 0.f);

    // yn = LN2(y)
    launch_ln(st, R.y, 64, R.yn, 64, P.ln2_g, P.ln2_b, Mr, 64, 64);

    // g = gelu(yn @ m1_w + m1_b)
    launch_gemm(st, 1, 2, R.yn, 64, 1, 0, R.pM1, 0, 0, 0, P.m1_b,
                nullptr, 0, nullptr, 0, 0, R.g, 64, 0,
                M, 64, 64, 1, EPI_GELU, 0.f);

    // h = y + (g @ m2_w + m2_b)
    launch_gemm(st, 1, 2, R.g, 64, 1, 0, R.pM2, 0, 0, 0, P.m2_b,
                nullptr, 0, R.y, 64, 0, R.h, 64, 0,
                M, 64, 64, 1, EPI_ADD, 0.f);
}

extern "C" void kernel_launch(void* const* d_in, const int* in_sizes, int n_in,
                              void* d_out, int out_size, void* d_ws, size_t ws_size,
                              hipStream_t stream)
{
    (void)in_sizes; (void)n_in; (void)out_size;
    const float* x = (const float*)d_in[0];
#define PF(i) ((const float*)d_in[(i)])
    PerfW P1 = { PF(1), PF(2), PF(3), PF(4), PF(5), PF(6), PF(7), PF(8),
                 PF(9), PF(10), PF(11), PF(12), PF(13) };
    PerfW P2 = { PF(14), PF(15), PF(16), PF(17), PF(18), PF(19), PF(20), PF(21),
                 PF(22), PF(23), PF(24), PF(25), PF(26) };
    const float* fproj_w = PF(27);
    const float* fproj_b = PF(28);
#undef PF

    // ---- workspace arena (floats) ----
    const long M1 = 64L * 3136;              // 200704 rows phase 1
    long off = 0;
    float* W = (float*)d_ws;
    auto take = [&](long n) { float* p = W + off; off += n; return p; };
    Regions R;
    R.X    = take(M1 * 152);                 // phase1: 200704x152; phase2: 50176x576 fits
    R.KQV  = take(M1 * 192);
    R.KP   = take(M1 * 32);
    R.QP   = take(M1 * 32);
    R.xdk  = take(M1);
    R.xdq  = take(M1);
    R.Dv   = take(M1);
    R.kps  = take(64L * 32);
    R.kptv = take(64L * 64 * 32);
    R.yatt = take(M1 * 64);
    R.y    = take(M1 * 64);
    R.yn   = take(M1 * 64);
    R.g    = take(M1 * 64);
    R.h    = take(M1 * 64);
    R.pKQV = take(576L * 192);               // max of 152*192 and 576*192
    R.pP   = take(64L * 64);
    R.pM1  = take(64L * 64);
    R.pM2  = take(64L * 64);
    float* pF = take(576L * 768);            // packed final projection
    if ((size_t)off * sizeof(float) > ws_size) return;

    // ---- phase 1: unfold(7,4,2) on NCHW (64,3,224,224) -> (200704, 147->152) ----
    {
        long total = 64L * 3136 * 147;
        unfold_kernel<<<(int)((total + 255) / 256), 256, 0, stream>>>(
            x, R.X, 64, 3, 224, 224, 7, 4, 2, 56, 56,
            3L * 224 * 224, 224L * 224, 1L, 152);
    }
    run_performer(stream, R, P1, 64, 3136, 147, 152);    // -> R.h (200704 x 64)

    // ---- phase 2: channel-last unfold(3,2,1) of (64, 56x56, 64) -> (50176, 576) ----
    {
        long total = 64L * 784 * 576;
        unfold_kernel<<<(int)((total + 255) / 256), 256, 0, stream>>>(
            R.h, R.X, 64, 64, 56, 56, 3, 2, 1, 28, 28,
            3136L * 64, 1L, 64L, 576);
    }
    run_performer(stream, R, P2, 64, 784, 576, 576);     // -> R.h (50176 x 64)

    // ---- phase 3: channel-last unfold(3,2,1) of (64, 28x28, 64) -> (12544, 576) ----
    {
        long total = 64L * 196 * 576;
        unfold_kernel<<<(int)((total + 255) / 256), 256, 0, stream>>>(
            R.h, R.X, 64, 64, 28, 28, 3, 2, 1, 14, 14,
            784L * 64, 1L, 64L, 576);
    }
    // final projection: (12544 x 576) @ (576 x 768) + b -> d_out
    launch_pack(stream, fproj_w, pF, 576, 768, 576);
    launch_gemm(stream, 1, 2, R.X, 576, 1, 0, pF, 0, 0, 0, fproj_b,
                nullptr, 0, nullptr, 0, 0, (float*)d_out, 768, 0,
                12544, 768, 576, 1, EPI_NONE, 0.f);
}